// Disout_53532472377614
// MI455X (gfx1250) — compile-verified
//
#include <hip/hip_runtime.h>
#include <hip/hip_bf16.h>
#include <cstdint>

// Disout (DropBlock variant), shapes fixed by the reference.
#define Bn 16
#define Wn 224
#define Hn 224
#define Cn 64
#define ROWE (Hn * Cn)                         // 14336 elements per (b,w) row
#define NTOT ((double)(16.0 * 224.0 * 224.0 * 64.0))
#define NELEMS ((size_t)Bn * Wn * Hn * Cn)     // 51,380,224
#define HMW ((size_t)Bn * Cn * Wn * 7)         // 1,605,632 packed words

// ---------------- CDNA5 Tensor Data Mover availability -----------------
#ifndef __has_builtin
#define __has_builtin(x) 0
#endif
#if __has_builtin(__builtin_amdgcn_tensor_load_to_lds) && \
    __has_builtin(__builtin_amdgcn_s_wait_tensorcnt)
#define HAVE_TDM 1
#else
#define HAVE_TDM 0
#endif

#if HAVE_TDM
#if __has_include(<hip/amd_detail/amd_gfx1250_TDM.h>)
#define TDM_ARGS6 1
#else
#define TDM_ARGS6 0
#endif
typedef unsigned int u32x4 __attribute__((ext_vector_type(4)));
typedef int i32x4 __attribute__((ext_vector_type(4)));
typedef int i32x8 __attribute__((ext_vector_type(8)));

// DMA one contiguous row of `nelem` 4-byte elements from global into LDS.
// D# group0: count=1, lds_addr, global_addr, type=2 ("image").
// D# group1: data_size=2 (4B), tensor_dim0 = tile_dim0 = nelem (1-D tile),
//            tensor_dim1 = tile_dim1 = 1, tensor_dim0_stride = nelem.
__device__ __forceinline__ void tdm_load_row(const void* gsrc, void* lds_dst,
                                             unsigned nelem) {
  unsigned long long ga = (unsigned long long)(uintptr_t)gsrc;
  unsigned lds = (unsigned)(uintptr_t)lds_dst;  // flat LDS addr low 32 bits
  u32x4 g0;
  g0[0] = 1u;                                        // count = 1 valid desc
  g0[1] = lds;                                       // lds_addr (bytes)
  g0[2] = (unsigned)(ga & 0xFFFFFFFFull);            // global_addr[31:0]
  g0[3] = (unsigned)((ga >> 32) & 0x1FFFFFFull)      // global_addr[56:32]
          | (2u << 30);                              // type = 2
  i32x8 g1;
  g1[0] = (int)(2u << 16);                           // data_size = 4B
  g1[1] = (int)((nelem & 0xFFFFu) << 16);            // tensor_dim0[15:0]
  g1[2] = (int)(((nelem >> 16) & 0xFFFFu)            // tensor_dim0[31:16]
          | (1u << 16));                             // tensor_dim1 lo = 1
  g1[3] = (int)((nelem & 0xFFFFu) << 16);            // tile_dim0 = nelem
  g1[4] = 1;                                         // tile_dim1 = 1
  g1[5] = (int)nelem;                                // tensor_dim0_stride
  g1[6] = 0;
  g1[7] = 0;
  i32x4 z4 = (i32x4)0;
#if TDM_ARGS6
  i32x8 z8 = (i32x8)0;
  __builtin_amdgcn_tensor_load_to_lds(g0, g1, z4, z4, z8, 0);
#else
  __builtin_amdgcn_tensor_load_to_lds(g0, g1, z4, z4, 0);
#endif
}
#endif  // HAVE_TDM

// ---------------- kernels ----------------------------------------------

__global__ void disout_init(double* s, double* q, unsigned long long* cnt) {
  if (threadIdx.x == 0 && blockIdx.x == 0) {
    *s = 0.0;
    *q = 0.0;
    *cnt = 0ull;
  }
}

// Global sum / sum-of-squares of x (for variance), f64 accumulation.
__global__ __launch_bounds__(256) void disout_var(const float4* __restrict__ x4,
                                                  double* sum, double* sumsq) {
  const int n4 = (int)(NELEMS / 4);
  double s = 0.0, q = 0.0;
  for (int i = blockIdx.x * blockDim.x + threadIdx.x; i < n4;
       i += gridDim.x * blockDim.x) {
    const float4 v = x4[i];
    s += (double)v.x + (double)v.y + (double)v.z + (double)v.w;
    q += (double)v.x * v.x + (double)v.y * v.y + (double)v.z * v.z +
         (double)v.w * v.w;
  }
  __shared__ double sh[256];
  sh[threadIdx.x] = s;
  __syncthreads();
  for (int o = 128; o > 0; o >>= 1) {
    if (threadIdx.x < (unsigned)o) sh[threadIdx.x] += sh[threadIdx.x + o];
    __syncthreads();
  }
  if (threadIdx.x == 0) atomicAdd(sum, sh[0]);
  __syncthreads();
  sh[threadIdx.x] = q;
  __syncthreads();
  for (int o = 128; o > 0; o >>= 1) {
    if (threadIdx.x < (unsigned)o) sh[threadIdx.x] += sh[threadIdx.x + o];
    __syncthreads();
  }
  if (threadIdx.x == 0) atomicAdd(sumsq, sh[0]);
}

// Pass 1: seed pattern bit-packed along H (224 bits -> 7 words per lane),
// then the horizontal 6-wide min(=AND) window done as shifted ANDs with
// 1-fill (SAME padding identity for AND).
// One workgroup (64 threads = 2 waves) per (b,w); lane owns channel c.
__global__ __launch_bounds__(64) void disout_seed_hpool(
    const float* __restrict__ rd, uint32_t* __restrict__ hm, float sdr) {
  const int w = blockIdx.x % Wn;
  const int b = blockIdx.x / Wn;
  const int c = threadIdx.x;
  const float* rowp = rd + (size_t)(b * Wn + w) * ROWE;
#if HAVE_TDM
  __shared__ float srow[ROWE];  // 57,344 B
  if (threadIdx.x == 0) tdm_load_row(rowp, srow, ROWE);
  __builtin_amdgcn_s_wait_tensorcnt(0);
  __syncthreads();
#endif
  const bool win_w = (w >= 3) && (w <= 221);
  const float base_in = 0.0f + (1.0f - sdr);   // interior: vbc = 1
  const float base_out = 1.0f + (1.0f - sdr);  // border:   vbc = 0
  uint32_t a[7];
#pragma unroll
  for (int j = 0; j < 7; ++j) {
    uint32_t m = 0;
#pragma unroll
    for (int k = 0; k < 32; ++k) {
      const int h = j * 32 + k;
#if HAVE_TDM
      const float r = srow[h * Cn + c];
#else
      const float r = rowp[h * Cn + c];
#endif
      const float base = (win_w && h >= 3 && h <= 221) ? base_in : base_out;
      if (base + r >= 1.0f) m |= (1u << k);
    }
    a[j] = m;
  }
  // hmin[h] = AND_{d=-2..3} seed[h+d], out-of-range -> 1
  uint32_t res[7];
#pragma unroll
  for (int j = 0; j < 7; ++j) {
    const uint32_t lo = (j > 0) ? a[j - 1] : 0xFFFFFFFFu;
    const uint32_t hi = (j < 6) ? a[j + 1] : 0xFFFFFFFFu;
    const uint32_t v = a[j];
    uint32_t r = v;
    r &= (v >> 1) | (hi << 31);
    r &= (v >> 2) | (hi << 30);
    r &= (v >> 3) | (hi << 29);
    r &= (v << 1) | (lo >> 31);
    r &= (v << 2) | (lo >> 30);
    res[j] = r;
  }
  uint32_t* dst = hm + (size_t)((b * Cn + c) * Wn + w) * 7;
#pragma unroll
  for (int j = 0; j < 7; ++j) dst[j] = res[j];
}

// Pass 2: vertical 6-wide AND across w rows (L2-resident, 6.4 MB total),
// plus popcount reduction for percent_ones.
__global__ __launch_bounds__(256) void disout_vpool(
    const uint32_t* __restrict__ hm, uint32_t* __restrict__ bp,
    unsigned long long* cnt) {
  const int t = blockIdx.x * blockDim.x + threadIdx.x;  // < 16*64*224
  const int w = t % Wn;
  const int s = t / Wn;  // b*64 + c
  const uint32_t* base = hm + (size_t)s * Wn * 7;
  uint32_t o[7] = {0xFFFFFFFFu, 0xFFFFFFFFu, 0xFFFFFFFFu, 0xFFFFFFFFu,
                   0xFFFFFFFFu, 0xFFFFFFFFu, 0xFFFFFFFFu};
#pragma unroll
  for (int d = -2; d <= 3; ++d) {
    const int wd = w + d;
    if (wd < 0 || wd >= Wn) continue;
    const uint32_t* row = base + wd * 7;
#pragma unroll
    for (int j = 0; j < 7; ++j) o[j] &= row[j];
  }
  uint32_t* dst = bp + (size_t)(s * Wn + w) * 7;
  int pc = 0;
#pragma unroll
  for (int j = 0; j < 7; ++j) {
    dst[j] = o[j];
    pc += __popc(o[j]);
  }
  __shared__ int sh[256];
  sh[threadIdx.x] = pc;
  __syncthreads();
  for (int o2 = 128; o2 > 0; o2 >>= 1) {
    if (threadIdx.x < (unsigned)o2) sh[threadIdx.x] += sh[threadIdx.x + o2];
    __syncthreads();
  }
  if (threadIdx.x == 0) atomicAdd(cnt, (unsigned long long)sh[0]);
}

// Final: out = (x*bp + 0.01*sqrt(var)*noise*(1-bp)) * (N/count).
// Same (b,w)-row-per-workgroup layout; x staged through LDS via TDM.
__global__ __launch_bounds__(64) void disout_final(
    const float* __restrict__ x, const float* __restrict__ nz,
    const uint32_t* __restrict__ bp, const double* __restrict__ sum,
    const double* __restrict__ sumsq, const unsigned long long* __restrict__ cnt,
    float* __restrict__ out) {
  const int w = blockIdx.x % Wn;
  const int b = blockIdx.x / Wn;
  const int c = threadIdx.x;
  const size_t rowoff = (size_t)(b * Wn + w) * ROWE;
#if HAVE_TDM
  __shared__ float srow[ROWE];  // 57,344 B
  if (threadIdx.x == 0) tdm_load_row(x + rowoff, srow, ROWE);
  __builtin_amdgcn_s_wait_tensorcnt(0);
  __syncthreads();
#endif
  const double Nd = NTOT;
  const double mean = *sum / Nd;
  double var = *sumsq / Nd - mean * mean;
  if (var < 0.0) var = 0.0;
  const float coef = (float)(0.01 * sqrt(var));            // ALPHA = 1
  const float scale = (float)(Nd / (double)(*cnt));        // 1/percent_ones
  const uint32_t* bprow = bp + (size_t)((b * Cn + c) * Wn + w) * 7;
#pragma unroll
  for (int j = 0; j < 7; ++j) {
    const uint32_t word = bprow[j];
#pragma unroll 8
    for (int k = 0; k < 32; ++k) {
      const int h = j * 32 + k;
      const size_t idx = rowoff + (size_t)(h * Cn + c);
#if HAVE_TDM
      const float xv = srow[h * Cn + c];
#else
      const float xv = x[idx];
#endif
      const float nv = nz[idx];
      const float p = (float)((word >> k) & 1u);
      out[idx] = (xv * p + coef * nv * (1.0f - p)) * scale;
    }
  }
}

// ---------------- launch -------------------------------------------------

extern "C" void kernel_launch(void* const* d_in, const int* in_sizes, int n_in,
                              void* d_out, int out_size, void* d_ws,
                              size_t ws_size, hipStream_t stream) {
  (void)in_sizes;
  (void)n_in;
  (void)out_size;
  (void)ws_size;
  const float* x = (const float*)d_in[0];
  const float* rd = (const float*)d_in[1];
  const float* nz = (const float*)d_in[2];
  float* out = (float*)d_out;

  char* ws = (char*)d_ws;
  double* d_sum = (double*)ws;
  double* d_sumsq = (double*)(ws + 8);
  unsigned long long* d_cnt = (unsigned long long*)(ws + 16);
  uint32_t* d_hm = (uint32_t*)(ws + 128);
  uint32_t* d_bp = d_hm + HMW;   // total ws use: ~12.3 MB

  constexpr double SDRd = 0.1 * (224.0 * 224.0) / 36.0 / (219.0 * 219.0);
  const float sdr = (float)SDRd;

  disout_init<<<1, 64, 0, stream>>>(d_sum, d_sumsq, d_cnt);
  disout_var<<<2048, 256, 0, stream>>>((const float4*)x, d_sum, d_sumsq);
  disout_seed_hpool<<<Bn * Wn, Cn, 0, stream>>>(rd, d_hm, sdr);
  disout_vpool<<<(Bn * Cn * Wn) / 256, 256, 0, stream>>>(d_hm, d_bp, d_cnt);
  disout_final<<<Bn * Wn, Cn, 0, stream>>>(x, nz, d_bp, d_sum, d_sumsq, d_cnt,
                                           out);
}